// CostSensitiveCrossEntropyLossN_90142773609053
// MI455X (gfx1250) — compile-verified
//
#include <hip/hip_runtime.h>
#include <hip/hip_bf16.h>
#include <math.h>

// ---------------------------------------------------------------------------
// CostSensitiveCrossEntropyLoss on MI455X (gfx1250, wave32)
// Memory-bound: 524 MB stream of `outputs` -> ~22.5 us floor @ 23.3 TB/s.
// No matmul exists in this op; the CDNA5 feature used is the async
// global->LDS pipe (global_load_async_to_lds_b128 / s_wait_asynccnt).
// Row pipeline: async global->LDS, LDS->VGPR (8x float4), two-pass softmax
// (max/argmax, then sum-exp with 4 independent accumulators).
// ---------------------------------------------------------------------------

#define TPB 256
#define WAVES_PER_BLOCK 8
#define KMAX 8                  // float4 chunks per lane
#define MAXQ (KMAX * 32)        // max float4s per row stageable (C <= 1024)

#if defined(__AMDGCN__) && __has_builtin(__builtin_amdgcn_global_load_async_to_lds_b128)
#define HAVE_ASYNC_LDS 1
#else
#define HAVE_ASYNC_LDS 0
#endif

#if HAVE_ASYNC_LDS
typedef int v4i_ty __attribute__((vector_size(16)));
#define AS_GLOBAL __attribute__((address_space(1)))
#define AS_LDS    __attribute__((address_space(3)))
#endif

__device__ __forceinline__ float wave_sum_f(float v) {
#pragma unroll
  for (int o = 16; o > 0; o >>= 1) v += __shfl_xor(v, o);
  return v;
}
__device__ __forceinline__ float wave_max_f(float v) {
#pragma unroll
  for (int o = 16; o > 0; o >>= 1) v = fmaxf(v, __shfl_xor(v, o));
  return v;
}

// ---------------------------------------------------------------------------
// Kernel A: per-row softmax stats. One wave per row.
//   -> atomicAdd(counts[t*C + argmax], 1)
//   -> sum_glp += x[t] - max - log(sum exp(x - max))
// ---------------------------------------------------------------------------
__global__ __launch_bounds__(TPB) void k_rowstats(
    const float* __restrict__ outputs, const int* __restrict__ targets,
    float* __restrict__ counts, float* __restrict__ sum_glp, int B, int C) {
#if HAVE_ASYNC_LDS
  __shared__ float4 lbuf[WAVES_PER_BLOCK][MAXQ];
#endif
  __shared__ float blockGlp;
  if (threadIdx.x == 0) blockGlp = 0.0f;
  __syncthreads();

  const int lane = threadIdx.x & 31;
  const int wave = threadIdx.x >> 5;
  const int gw   = blockIdx.x * WAVES_PER_BLOCK + wave;
  const int nw   = gridDim.x * WAVES_PER_BLOCK;
  const int Q    = C >> 2; // float4s per row
  const bool fast = ((C & 3) == 0) && (Q <= MAXQ);

  float acc = 0.0f;

  for (int row = gw; row < B; row += nw) {
    const float* rowf = outputs + (size_t)row * (size_t)C;
    float m;  // row max
    int   bi; // first index attaining it
    float s;  // sum exp(x - m)

    if (fast) {
      const float4* rowp = (const float4*)rowf;
      float4 rv[KMAX];

#if HAVE_ASYNC_LDS
      // Stage the row into this wave's LDS slab via the CDNA5 async pipe
      // (tracked by ASYNCcnt; no VGPR involvement on the inbound path).
#pragma unroll
      for (int k = 0; k < KMAX; ++k) {
        int idx = lane + (k << 5);
        if (idx < Q) {
          __builtin_amdgcn_global_load_async_to_lds_b128(
              (AS_GLOBAL v4i_ty*)(rowp + idx),
              (AS_LDS v4i_ty*)&lbuf[wave][idx],
              /*offset=*/0, /*cpol=*/0);
        }
      }
#if __has_builtin(__builtin_amdgcn_s_wait_asynccnt)
      __builtin_amdgcn_s_wait_asynccnt(0);
#else
      asm volatile("s_wait_asynccnt 0" ::: "memory");
#endif
      // LDS -> registers (single ds pass; both softmax passes run from VGPRs)
#pragma unroll
      for (int k = 0; k < KMAX; ++k) {
        int idx = lane + (k << 5);
        if (idx < Q) rv[k] = lbuf[wave][idx];
      }
#else
#pragma unroll
      for (int k = 0; k < KMAX; ++k) {
        int idx = lane + (k << 5);
        if (idx < Q) rv[k] = rowp[idx];
      }
#endif
      // ---- pass 1: max + first-occurrence argmax (register data) ----
      m = -3.402823466e38f;
      bi = 0;
      auto updm = [&](float x, int c) {
        if (x > m) { m = x; bi = c; } // tie keeps earlier (smaller) index
      };
#pragma unroll
      for (int k = 0; k < KMAX; ++k) {
        int idx = lane + (k << 5);
        if (idx < Q) {
          int c = idx << 2;
          updm(rv[k].x, c); updm(rv[k].y, c + 1);
          updm(rv[k].z, c + 2); updm(rv[k].w, c + 3);
        }
      }
      // wave32 combine (value desc, index asc on ties)
#pragma unroll
      for (int o = 16; o > 0; o >>= 1) {
        float m2 = __shfl_xor(m, o);
        int   b2 = __shfl_xor(bi, o);
        if (m2 > m || (m2 == m && b2 < bi)) { m = m2; bi = b2; }
      }

      // ---- pass 2: sum exp(x - M), 4 independent chains ----
      float s0 = 0.f, s1 = 0.f, s2 = 0.f, s3 = 0.f;
#pragma unroll
      for (int k = 0; k < KMAX; ++k) {
        int idx = lane + (k << 5);
        if (idx < Q) {
          s0 += __expf(rv[k].x - m);
          s1 += __expf(rv[k].y - m);
          s2 += __expf(rv[k].z - m);
          s3 += __expf(rv[k].w - m);
        }
      }
      s = wave_sum_f((s0 + s1) + (s2 + s3));
    } else {
      // generic path (any C): two passes straight from global
      m = -3.402823466e38f;
      bi = 0;
      for (int c = lane; c < C; c += 32) {
        float x = rowf[c];
        if (x > m) { m = x; bi = c; }
      }
#pragma unroll
      for (int o = 16; o > 0; o >>= 1) {
        float m2 = __shfl_xor(m, o);
        int   b2 = __shfl_xor(bi, o);
        if (m2 > m || (m2 == m && b2 < bi)) { m = m2; bi = b2; }
      }
      float ls = 0.f;
      for (int c = lane; c < C; c += 32) ls += __expf(rowf[c] - m);
      s = wave_sum_f(ls);
    }

    if (lane == 0) {
      int t = targets[row];
      float xt = rowf[t];
      acc += xt - m - __logf(s);                            // log-prob of true class
      atomicAdd(&counts[(size_t)t * (size_t)C + bi], 1.0f); // confusion scatter
    }
  }

  if (lane == 0) atomicAdd(&blockGlp, acc); // LDS atomic
  __syncthreads();
  if (threadIdx.x == 0) atomicAdd(sum_glp, blockGlp);
}

// ---------------------------------------------------------------------------
// Kernel B: v = (cost + count)^0.25, zero diagonal, global max.
// One block per matrix row (no divisions). atomicMax on float bits is
// order-correct for non-negative floats.
// ---------------------------------------------------------------------------
__global__ __launch_bounds__(TPB) void k_powmax(
    const float* __restrict__ cost, const float* __restrict__ counts,
    unsigned int* __restrict__ maxbits, int C) {
  const int r = blockIdx.x;
  const float* rowc = cost + (size_t)r * C;
  const float* rowk = counts + (size_t)r * C;
  float lmax = 0.0f;
  for (int c = threadIdx.x; c < C; c += blockDim.x) {
    float v = (c == r) ? 0.0f : sqrtf(sqrtf(rowc[c] + rowk[c]));
    lmax = fmaxf(lmax, v);
  }
  lmax = wave_max_f(lmax);
  __shared__ float red[WAVES_PER_BLOCK];
  if ((threadIdx.x & 31) == 0) red[threadIdx.x >> 5] = lmax;
  __syncthreads();
  if (threadIdx.x < WAVES_PER_BLOCK) {
    float v = red[threadIdx.x];
#pragma unroll
    for (int o = WAVES_PER_BLOCK / 2; o > 0; o >>= 1) v = fmaxf(v, __shfl_xor(v, o));
    if (threadIdx.x == 0) atomicMax(maxbits, __float_as_uint(v));
  }
}

// ---------------------------------------------------------------------------
// Kernel C: Sum(gc) = sum over cells of count * clip(v * 2/max, 1, 2).
// Diagonal cells with count>0 give v=0 -> clipped to 1 (matches reference).
// ---------------------------------------------------------------------------
__global__ __launch_bounds__(TPB) void k_gcsum(
    const float* __restrict__ cost, const float* __restrict__ counts,
    const unsigned int* __restrict__ maxbits, float* __restrict__ gc_sum, int C) {
  const int r = blockIdx.x;
  const float scale = 2.0f / __uint_as_float(*maxbits);
  const float* rowc = cost + (size_t)r * C;
  const float* rowk = counts + (size_t)r * C;
  float lsum = 0.0f;
  for (int c = threadIdx.x; c < C; c += blockDim.x) {
    float cnt = rowk[c];
    if (cnt != 0.0f) {
      float v = (c == r) ? 0.0f : sqrtf(sqrtf(rowc[c] + cnt));
      v = fminf(fmaxf(v * scale, 1.0f), 2.0f);
      lsum += cnt * v;
    }
  }
  lsum = wave_sum_f(lsum);
  __shared__ float red[WAVES_PER_BLOCK];
  if ((threadIdx.x & 31) == 0) red[threadIdx.x >> 5] = lsum;
  __syncthreads();
  if (threadIdx.x < WAVES_PER_BLOCK) {
    float v = red[threadIdx.x];
#pragma unroll
    for (int o = WAVES_PER_BLOCK / 2; o > 0; o >>= 1) v += __shfl_xor(v, o);
    if (threadIdx.x == 0) atomicAdd(gc_sum, v);
  }
}

// ---------------------------------------------------------------------------
// Kernel D: loss = -(mean(glp) * mean(gc))
// ---------------------------------------------------------------------------
__global__ void k_final(const float* __restrict__ sum_glp,
                        const float* __restrict__ gc_sum,
                        float* __restrict__ out, float invB) {
  if (threadIdx.x == 0 && blockIdx.x == 0)
    out[0] = -((*sum_glp) * invB) * ((*gc_sum) * invB);
}

// ---------------------------------------------------------------------------
extern "C" void kernel_launch(void* const* d_in, const int* in_sizes, int n_in,
                              void* d_out, int out_size, void* d_ws, size_t ws_size,
                              hipStream_t stream) {
  const float* outputs = (const float*)d_in[0]; // [B, C] fp32
  const int*   targets = (const int*)d_in[1];   // [B] int
  const float* cost    = (const float*)d_in[2]; // [C, C] fp32

  const int B = in_sizes[1];
  const int C = (int)(in_sizes[0] / (size_t)B);

  // workspace: counts[C*C] then {sum_glp, gc_sum, maxbits, pad}
  float* counts = (float*)d_ws;
  float* scal   = counts + (size_t)C * (size_t)C;
  (void)hipMemsetAsync(d_ws, 0, ((size_t)C * (size_t)C + 4) * sizeof(float), stream);

  int blocksA = (B + WAVES_PER_BLOCK - 1) / WAVES_PER_BLOCK;
  if (blocksA > 2048) blocksA = 2048;
  k_rowstats<<<blocksA, TPB, 0, stream>>>(outputs, targets, counts, scal + 0, B, C);

  k_powmax<<<C, TPB, 0, stream>>>(cost, counts, (unsigned int*)(scal + 2), C);
  k_gcsum<<<C, TPB, 0, stream>>>(cost, counts, (const unsigned int*)(scal + 2),
                                 scal + 1, C);
  k_final<<<1, 32, 0, stream>>>(scal + 0, scal + 1, (float*)d_out, 1.0f / (float)B);
}